// NFFT_54065048322177
// MI455X (gfx1250) — compile-verified
//
#include <hip/hip_runtime.h>
#include <stdint.h>

// ---------------- problem constants (from reference) ----------------
#define NMODES 4096      // N
#define MTAPS  4         // m
#define NGRID  8192      // n = 2N
#define BATCH  16
#define NPTS   131072

#define PI_F  3.14159265358979323846f
#define BKB   (1.5f * PI_F)              // (2 - 1/sigma)*pi, sigma = 2

typedef float    v2f  __attribute__((ext_vector_type(2)));
typedef float    v8f  __attribute__((ext_vector_type(8)));
typedef uint32_t u32x4 __attribute__((ext_vector_type(4)));
typedef int      i32x4 __attribute__((ext_vector_type(4)));
typedef int      i32x8 __attribute__((ext_vector_type(8)));

// ---------------- Tensor Data Mover: 2D tile global->LDS ----------------
// Builds a D# per CDNA5 ISA and issues TENSOR_LOAD_TO_LDS.
// dims/strides are in data_size units. dsize_code: 0=1B,1=2B,2=4B,3=8B.
// This toolchain uses the 6-arg builtin: (g0, g1, g2, g3, extra, cpol).
__device__ __forceinline__ void tdm_load_2d(uint32_t lds_byte_addr, const void* gptr,
                                            uint32_t tensor_d0, uint32_t tensor_d1,
                                            uint32_t tile_d0,   uint32_t tile_d1,
                                            uint64_t stride0,   uint32_t dsize_code,
                                            uint32_t pad_en, uint32_t pad_interval,
                                            uint32_t pad_amount)
{
    uint64_t ga = (uint64_t)(uintptr_t)gptr;
    u32x4 g0;
    g0[0] = 1u;                                  // count=1, user descriptor
    g0[1] = lds_byte_addr;                       // lds_addr [63:32]
    g0[2] = (uint32_t)ga;                        // global_addr [95:64]
    g0[3] = ((uint32_t)(ga >> 32) & 0x01ffffffu) // global_addr [120:96]
          | (2u << 30);                          // type = 2 ("image")
    i32x8 g1;
    g1[0] = (int)((dsize_code << 16) | (pad_en << 20) |
                  (pad_interval << 22) | (pad_amount << 25)); // wg_mask=0
    g1[1] = (int)((tensor_d0 & 0xffffu) << 16);               // tensor_dim0[15:0] @ bit48
    g1[2] = (int)((tensor_d0 >> 16) | ((tensor_d1 & 0xffffu) << 16));
    g1[3] = (int)((tensor_d1 >> 16) | (tile_d0 << 16));
    g1[4] = (int)(tile_d1);                                   // tile_dim2 = 0
    g1[5] = (int)(uint32_t)stride0;                           // tensor_dim0_stride[31:0]
    g1[6] = (int)((uint32_t)(stride0 >> 32) & 0xffffu);       // stride0 hi, stride1=0
    g1[7] = 0;
    i32x4 z4 = {0, 0, 0, 0};                                  // groups 2/3 zero (<=2D)
    i32x8 z8 = {0, 0, 0, 0, 0, 0, 0, 0};                      // unused extra group
    __builtin_amdgcn_tensor_load_to_lds(g0, g1, z4, z4, z8, 0);
}

// ---------------- Kaiser-Bessel helpers ----------------
__device__ __forceinline__ float bessel_i0f(float x)
{
    float ax = fabsf(x);
    if (ax < 3.75f) {
        float t = x / 3.75f; t *= t;
        return 1.0f + t*(3.5156229f + t*(3.0899424f + t*(1.2067492f +
                     t*(0.2659732f + t*(0.0360768f + t*0.0045813f)))));
    } else {
        float t = 3.75f / ax;
        float p = 0.39894228f + t*(0.01328592f + t*(0.00225319f + t*(-0.00157565f +
                  t*(0.00916281f + t*(-0.02057706f + t*(0.02635537f +
                  t*(-0.01647633f + t*0.00392377f)))))));
        return p * expf(ax) * rsqrtf(ax);
    }
}

// ---------------- Kernel 0: deconvolve f_hat / phi_hat ----------------
__global__ __launch_bounds__(256) void nufft_deconv(const float* __restrict__ fr,
                                                    const float* __restrict__ fi,
                                                    float* __restrict__ cre,
                                                    float* __restrict__ cim)
{
    int t = blockIdx.x * blockDim.x + threadIdx.x;   // [0, 65536)
    int s = t & (NMODES - 1);
    // phi_hat[s] = I0(m * sqrt(b^2 - (2*pi*(s - N/2)/n)^2))
    float w   = (float)(s - NMODES / 2) * (PI_F / (float)(NGRID / 2));
    float arg = (float)MTAPS * sqrtf(BKB * BKB - w * w);
    float inv = 1.0f / bessel_i0f(arg);
    cre[t] = fr[t] * inv;
    cim[t] = fi[t] * inv;
}

// ---------------- Kernel 1: DFT as complex WMMA GEMM ----------------
// h[b][j] = sum_s c[b][s] * exp(-2*pi*i*(s - N/2)*j / n),  j in [0, n)
// One wave per 16-wide j-tile; 8 waves (8 tiles) per block.
// c is streamed through double-buffered LDS (16 chunks of 256 columns) via
// TDM; the in-flight pair for chunk ch+1 overlaps WMMA work on chunk ch
// (TENSORcnt completes in order, so s_wait_tensorcnt(2) retires chunk ch).
// LDS pitch 260 floats (TDM pad: 4 dwords every 256) -> the 16 A-tile rows
// land in 16 distinct banks.
#define CHUNK 256
#define PITCH 260

__global__ __launch_bounds__(256) void nufft_dft_wmma(const float* __restrict__ cre,
                                                      const float* __restrict__ cim,
                                                      float2* __restrict__ h)
{
    __shared__ float sre[2][BATCH * PITCH];
    __shared__ float sim[2][BATCH * PITCH];

    const int tid  = threadIdx.x;
    const int wave = tid >> 5;
    const int lane = tid & 31;
    const int row  = lane & 15;                 // A: batch row / B: j column
    const int koff = (lane < 16) ? 0 : 2;       // 32-bit 16x4 striping (ISA 7.12.2)
    const int jbase = (blockIdx.x * 8 + wave) * 16;
    const int j     = jbase + row;
    const float w0  = 6.283185307179586f / (float)NGRID;

    // constant per-lane rotator: phase advance for t -> t+4 is (4j mod n)*w0
    const int   dp    = (4 * j) & (NGRID - 1);
    const float delta = (float)dp * w0;
    const float rc = __cosf(delta);
    const float rs = __sinf(delta);

    v8f acc_re = {0.f,0.f,0.f,0.f,0.f,0.f,0.f,0.f};
    v8f acc_im = {0.f,0.f,0.f,0.f,0.f,0.f,0.f,0.f};

    if (tid == 0) {                                  // prologue: chunk 0 -> buf 0
        tdm_load_2d((uint32_t)(uintptr_t)(void*)sre[0], cre, NMODES, BATCH,
                    CHUNK, BATCH, NMODES, 2, 1, 7, 3);
        tdm_load_2d((uint32_t)(uintptr_t)(void*)sim[0], cim, NMODES, BATCH,
                    CHUNK, BATCH, NMODES, 2, 1, 7, 3);
    }

    for (int ch = 0; ch < NMODES / CHUNK; ++ch) {
        const int buf = ch & 1;
        if (tid == 0) {
            if (ch + 1 < NMODES / CHUNK) {           // prefetch next chunk
                const int nb = (ch + 1) & 1;
                tdm_load_2d((uint32_t)(uintptr_t)(void*)sre[nb],
                            cre + (ch + 1) * CHUNK, NMODES, BATCH,
                            CHUNK, BATCH, NMODES, 2, 1, 7, 3);
                tdm_load_2d((uint32_t)(uintptr_t)(void*)sim[nb],
                            cim + (ch + 1) * CHUNK, NMODES, BATCH,
                            CHUNK, BATCH, NMODES, 2, 1, 7, 3);
                __builtin_amdgcn_s_wait_tensorcnt(2);   // chunk ch pair retired
            } else {
                __builtin_amdgcn_s_wait_tensorcnt(0);   // last chunk retired
            }
        }
        __syncthreads();                             // buf[ch&1] visible to all waves

        // exact integer re-seed of the twiddle pair at each chunk boundary
        const int tg = ch * CHUNK - NMODES / 2 + koff;   // t = s - N/2 for vgpr row 0
        int p0 = (tg * j)       & (NGRID - 1);
        int p1 = ((tg + 1) * j) & (NGRID - 1);
        float c0, s0, c1, s1;
        __sincosf((float)p0 * w0, &s0, &c0);
        __sincosf((float)p1 * w0, &s1, &c1);

        const float* bre = &sre[buf][row * PITCH];
        const float* bim = &sim[buf][row * PITCH];
        for (int k = 0; k < CHUNK / 4; ++k) {
            int sl = 4 * k + koff;
            const float2 ar2 = *(const float2*)&bre[sl];
            const float2 ai2 = *(const float2*)&bim[sl];
            v2f a_re = {ar2.x, ar2.y};
            v2f a_im = {ai2.x, ai2.y};
            v2f b_c  = { c0,  c1};
            v2f b_s  = { s0,  s1};
            v2f b_sn = {-s0, -s1};

            // h = c * e^{-i theta}: re += cr*cos + ci*sin ; im += ci*cos - cr*sin
            acc_re = __builtin_amdgcn_wmma_f32_16x16x4_f32(false, a_re, false, b_c,
                                                           (short)0, acc_re, false, false);
            acc_re = __builtin_amdgcn_wmma_f32_16x16x4_f32(false, a_im, false, b_s,
                                                           (short)0, acc_re, false, false);
            acc_im = __builtin_amdgcn_wmma_f32_16x16x4_f32(false, a_im, false, b_c,
                                                           (short)0, acc_im, false, false);
            acc_im = __builtin_amdgcn_wmma_f32_16x16x4_f32(false, a_re, false, b_sn,
                                                           (short)0, acc_im, false, false);

            // rotate both twiddle rows by delta (8 FMAs, no transcendentals)
            float nc0 = c0 * rc - s0 * rs;  s0 = fmaf(s0, rc,  c0 * rs);  c0 = nc0;
            float nc1 = c1 * rc - s1 * rs;  s1 = fmaf(s1, rc,  c1 * rs);  c1 = nc1;
        }
        __syncthreads();                             // all done reading buf[ch&1]
    }

    // C/D layout: VGPR v -> row (lane<16 ? v : v+8), col = lane&15
    #pragma unroll
    for (int v = 0; v < 8; ++v) {
        int r = (lane < 16) ? v : v + 8;
        h[r * NGRID + j] = make_float2(acc_re[v], acc_im[v]);
    }
}

// ---------------- Kernel 2: 8-tap KB interpolation (gather from LDS) ----------------
__global__ __launch_bounds__(256) void nufft_interp(const float* __restrict__ x,
                                                    const float2* __restrict__ h,
                                                    float2* __restrict__ out)
{
    __shared__ float2 sh[NGRID];                 // 64 KB: full grid for one batch
    const int b     = blockIdx.x >> 6;           // 64 point-chunks per batch
    const int chunk = blockIdx.x & 63;

    if (threadIdx.x == 0) {
        // 1D tile: 8192 x 8B elements, global -> LDS
        tdm_load_2d((uint32_t)(uintptr_t)(void*)sh, h + b * NGRID,
                    NGRID, 1, NGRID, 1, NGRID, /*8B*/3, 0, 0, 0);
        __builtin_amdgcn_s_wait_tensorcnt(0);
    }
    __syncthreads();

    const int pbase = chunk * 2048 + threadIdx.x;
    #pragma unroll 4
    for (int it = 0; it < 8; ++it) {
        int   p  = pbase + it * 256;
        float xv = x[b * NPTS + p];
        float nx = (float)NGRID * xv;
        int   i0 = (int)ceilf(nx) - MTAPS;
        float ar = 0.f, ai = 0.f;
        #pragma unroll
        for (int l = 0; l < 2 * MTAPS; ++l) {
            int   idx = (i0 + l) & (NGRID - 1);  // periodic wrap (shift-free form)
            float d   = nx - (float)(i0 + l);
            float a2  = (float)(MTAPS * MTAPS) - d * d;
            float w   = 0.0f;
            if (a2 > 0.0f) {
                float ag = sqrtf(a2);
                w = sinhf(BKB * ag) / (ag * PI_F);   // NaN cases (a2<=0) -> 0
            }
            float2 hv = sh[idx];
            ar = fmaf(w, hv.x, ar);
            ai = fmaf(w, hv.y, ai);
        }
        out[b * NPTS + p] = make_float2(ar, ai);
    }
}

// ---------------- host-side launch ----------------
extern "C" void kernel_launch(void* const* d_in, const int* in_sizes, int n_in,
                              void* d_out, int out_size, void* d_ws, size_t ws_size,
                              hipStream_t stream)
{
    (void)in_sizes; (void)n_in; (void)out_size; (void)ws_size;
    const float* x  = (const float*)d_in[0];   // [16, 131072]
    const float* fr = (const float*)d_in[1];   // [16, 4096]
    const float* fi = (const float*)d_in[2];   // [16, 4096]

    float*  ws  = (float*)d_ws;
    float*  cre = ws;                           // 65536 f32  (256 KB)
    float*  cim = ws + BATCH * NMODES;          // 65536 f32  (256 KB)
    float2* h   = (float2*)(ws + 2 * BATCH * NMODES); // 131072 float2 (1 MB)

    nufft_deconv  <<<(BATCH * NMODES) / 256, 256, 0, stream>>>(fr, fi, cre, cim);
    nufft_dft_wmma<<<(NGRID / 16) / 8,       256, 0, stream>>>(cre, cim, h);
    nufft_interp  <<<BATCH * 64,             256, 0, stream>>>(x, h, (float2*)d_out);
}